// GNNEncoder_71854802862395
// MI455X (gfx1250) — compile-verified
//
#include <hip/hip_runtime.h>

// ---------------------------------------------------------------------------
// GNN encoder (bipartite GraphSAGE, 2 layers) for MI455X / gfx1250.
//
// Roofline: GEMMs total ~12.8 GFLOP (negligible at WMMA rates). The bound is
// edge gather/scatter: 4 aggregations x 640K edges x 512B ~= 2.6 GB of
// irregular traffic -> ~115us @ 23.3 TB/s, with node tensors (10-26 MB)
// L2-resident (192 MB L2). So: simple atomic scatter for aggregation, WMMA
// (V_WMMA_F32_16X16X4_F32, full fp32 precision) for every GEMM, and fusion
// to remove extra full-tensor passes:
//   - mean-divide fused into the LDS A-panel staging of the SAGE linear
//   - agg@Wl + bl + x_dst@Wr fused into ONE kernel (two WMMA K-loops, one
//     accumulator) -> no C round-trip between the two linears.
// ---------------------------------------------------------------------------

typedef __attribute__((ext_vector_type(2))) float v2f;
typedef __attribute__((ext_vector_type(8))) float v8f;

#define NU      50000
#define NE      20000
#define NEDGE   640000
#define DH      128
#define BN_EPS  1e-5f

// ---------------------------------------------------------------------------
// WMMA fragment conventions (wave32, 16x16 f32 tile per wave):
//  A (16x4): lane l<16 holds row M=l, K=k..k+1; lane l+16 holds M=l, K=k+2..k+3
//  B (4x16): lane l<16 holds col N=l, K=k..k+1; lane l+16 holds N=l, K=k+2..k+3
//  C/D     : VGPR r, lane l -> M = r + 8*(l>=16), N = l%16
// Block = 256 threads = 8 waves covering cols [0,128); grid.x = N/16.
// N is always a multiple of 16 (50000/16=3125, 20000/16=1250) -> EXEC all-1s.
// ---------------------------------------------------------------------------

__device__ __forceinline__ v8f wmma_loop_k(const float* __restrict__ Wbase,
                                           const float* __restrict__ lds_panel,
                                           int ldsK, int K, int ln, int g, v8f acc)
{
    const float* wp = Wbase + ln;   // caller pre-offsets by col0
    for (int k = 0; k < K; k += 4) {
        v2f a = *(const v2f*)&lds_panel[ln * ldsK + k + 2 * g];   // ds_load_b64
        v2f b;
        b.x = wp[(size_t)(k + 2 * g    ) * DH];
        b.y = wp[(size_t)(k + 2 * g + 1) * DH];
        acc = __builtin_amdgcn_wmma_f32_16x16x4_f32(
                  false, a, false, b, (short)0, acc, false, false);
    }
    return acc;
}

// ---------------------------------------------------------------------------
// Generic projection GEMM: C[N x 128] = A[N x K] @ W[K x 128] + bias
// K in {64, 256} (power of two).
// ---------------------------------------------------------------------------
__global__ __launch_bounds__(256)
void gemm16_wmma_f32(const float* __restrict__ A, const float* __restrict__ W,
                     const float* __restrict__ bias, float* __restrict__ C, int K)
{
    __shared__ float As[16 * 260];          // K <= 256, padded stride K+4
    const int ldsK = K + 4;
    const int row0 = blockIdx.x * 16;

    const int klog = 31 - __clz(K);
    for (int idx = threadIdx.x; idx < (16 << klog); idx += 256) {
        int r = idx >> klog;
        int c = idx & (K - 1);
        As[r * ldsK + c] = A[(size_t)(row0 + r) * K + c];
    }
    __syncthreads();

    const int lane = threadIdx.x & 31;
    const int ln   = lane & 15;
    const int g    = lane >> 4;
    const int col0 = (threadIdx.x >> 5) * 16;

    v8f acc;
    float bv = bias[col0 + ln];
#pragma unroll
    for (int r = 0; r < 8; ++r) acc[r] = bv;

    acc = wmma_loop_k(W + col0, As, ldsK, K, ln, g, acc);

#pragma unroll
    for (int r = 0; r < 8; ++r)
        C[(size_t)(row0 + r + 8 * g) * DH + col0 + ln] = acc[r];
}

// ---------------------------------------------------------------------------
// Fused SAGE linear: out = (agg/max(deg,1)) @ Wl + bl + xdst @ Wr
// K = DH = 128 for both. Mean-divide fused into panel staging.
// ---------------------------------------------------------------------------
__global__ __launch_bounds__(256)
void sage_linear_wmma(const float* __restrict__ agg, const float* __restrict__ deg,
                      const float* __restrict__ xdst,
                      const float* __restrict__ Wl, const float* __restrict__ bl,
                      const float* __restrict__ Wr, float* __restrict__ out)
{
    constexpr int ldsK = DH + 4;            // 132, conflict-free across M stripe
    __shared__ float P0[16 * ldsK];         // mean-aggregated neighbor panel
    __shared__ float P1[16 * ldsK];         // destination self-feature panel
    const int row0 = blockIdx.x * 16;

    for (int idx = threadIdx.x; idx < 16 * DH; idx += 256) {
        int r = idx >> 7;
        int c = idx & (DH - 1);
        size_t off = (size_t)(row0 + r) * DH + c;
        float inv = 1.0f / fmaxf(deg[row0 + r], 1.0f);
        P0[r * ldsK + c] = agg[off] * inv;  // fused mean
        P1[r * ldsK + c] = xdst[off];
    }
    __syncthreads();

    const int lane = threadIdx.x & 31;
    const int ln   = lane & 15;
    const int g    = lane >> 4;
    const int col0 = (threadIdx.x >> 5) * 16;

    v8f acc;
    float bv = bl[col0 + ln];
#pragma unroll
    for (int r = 0; r < 8; ++r) acc[r] = bv;

    acc = wmma_loop_k(Wl + col0, P0, ldsK, DH, ln, g, acc);   // agg-mean @ Wl
    acc = wmma_loop_k(Wr + col0, P1, ldsK, DH, ln, g, acc);   // xdst @ Wr

#pragma unroll
    for (int r = 0; r < 8; ++r)
        out[(size_t)(row0 + r + 8 * g) * DH + col0 + ln] = acc[r];
}

// ---------------------------------------------------------------------------
// Edge scatter-add: agg[dst[e]] += x_src[src[e]], deg[dst[e]] += 1.
// One thread per (edge, 16B chunk): 32 float4 chunks cover D=128.
// ---------------------------------------------------------------------------
__global__ __launch_bounds__(256)
void scatter_add(const float* __restrict__ xsrc, const int* __restrict__ srci,
                 const int* __restrict__ dsti, float* __restrict__ agg,
                 float* __restrict__ deg, int nedge)
{
    long idx = (long)blockIdx.x * blockDim.x + threadIdx.x;
    if (idx >= (long)nedge * 32) return;
    int e = (int)(idx >> 5);
    int c = (int)(idx & 31) << 2;
    int s = srci[e];
    int d = dsti[e];
    float4 v = *(const float4*)(xsrc + (size_t)s * DH + c);
    float* out = agg + (size_t)d * DH + c;
    atomicAdd(out + 0, v.x);
    atomicAdd(out + 1, v.y);
    atomicAdd(out + 2, v.z);
    atomicAdd(out + 3, v.w);
    if (c == 0) atomicAdd(deg + d, 1.0f);
}

// per-column sum / sum-of-squares: per-thread partials, then atomic combine
__global__ __launch_bounds__(128)
void bn_stats(const float* __restrict__ x, float* __restrict__ stats,
              int n, int rows_per_block)
{
    int c  = threadIdx.x;                   // column 0..127
    int r0 = blockIdx.x * rows_per_block;
    int r1 = min(r0 + rows_per_block, n);
    float s = 0.0f, s2 = 0.0f;
    for (int r = r0; r < r1; ++r) {
        float v = x[(size_t)r * DH + c];
        s  += v;
        s2 += v * v;
    }
    atomicAdd(&stats[c], s);
    atomicAdd(&stats[DH + c], s2);
}

// y = relu(gamma * (x - mean) * rsqrt(var + eps) + beta)
__global__ __launch_bounds__(256)
void bn_apply_relu(const float* __restrict__ x, const float* __restrict__ stats,
                   const float* __restrict__ gamma, const float* __restrict__ beta,
                   float* __restrict__ y, int n)
{
    long idx = (long)blockIdx.x * blockDim.x + threadIdx.x;
    if (idx >= (long)n * DH) return;
    int c = (int)(idx & (DH - 1));
    float inv_n = 1.0f / (float)n;
    float mean  = stats[c] * inv_n;
    float var   = stats[DH + c] * inv_n - mean * mean;
    float v = gamma[c] * (x[idx] - mean) * rsqrtf(var + BN_EPS) + beta[c];
    y[idx] = fmaxf(v, 0.0f);
}

// ---------------------------------------------------------------------------
// Host-side orchestration
// ---------------------------------------------------------------------------
extern "C" void kernel_launch(void* const* d_in, const int* in_sizes, int n_in,
                              void* d_out, int out_size, void* d_ws, size_t ws_size,
                              hipStream_t stream)
{
    const float* x_user       = (const float*)d_in[0];   // [50000 x 256]
    const float* x_event      = (const float*)d_in[1];   // [20000 x 64]
    const int*   edge         = (const int*)  d_in[2];   // [2 x 640000]
    const float* user_proj_W  = (const float*)d_in[3];
    const float* user_proj_b  = (const float*)d_in[4];
    const float* event_proj_W = (const float*)d_in[5];
    const float* event_proj_b = (const float*)d_in[6];
    const float* user_Wl      = (const float*)d_in[7];
    const float* user_bl      = (const float*)d_in[8];
    const float* user_Wr      = (const float*)d_in[9];
    const float* user_gamma   = (const float*)d_in[10];
    const float* user_beta    = (const float*)d_in[11];
    const float* event_Wl     = (const float*)d_in[12];
    const float* event_bl     = (const float*)d_in[13];
    const float* event_Wr     = (const float*)d_in[14];
    const float* event_gamma  = (const float*)d_in[15];
    const float* event_beta   = (const float*)d_in[16];

    const int* u_idx = edge;            // edge_index[0]
    const int* e_idx = edge + NEDGE;    // edge_index[1]

    // live node embeddings sit directly in d_out (final BN+ReLU writes there)
    float* xu = (float*)d_out;                       // [50000 x 128]
    float* xe = xu + (size_t)NU * DH;                // [20000 x 128]

    // workspace layout
    float* agg   = (float*)d_ws;                     // up to [50000 x 128]
    float* xnew  = agg  + (size_t)NU * DH;           // up to [50000 x 128]
    float* deg   = xnew + (size_t)NU * DH;           // up to [50000]
    float* stats = deg  + NU;                        // [256] (sum, sumsq)

    const int scat_blocks = (int)(((long)NEDGE * 32 + 255) / 256);

    // ---- input projections (WMMA GEMMs) ----
    gemm16_wmma_f32<<<NU / 16, 256, 0, stream>>>(x_user,  user_proj_W,  user_proj_b,  xu, 256);
    gemm16_wmma_f32<<<NE / 16, 256, 0, stream>>>(x_event, event_proj_W, event_proj_b, xe,  64);

    for (int layer = 0; layer < 2; ++layer) {
        const size_t off_w = (size_t)layer * DH * DH;
        const size_t off_v = (size_t)layer * DH;

        // ---------------- user update: messages event -> user ----------------
        hipMemsetAsync(agg, 0, (size_t)NU * DH * sizeof(float), stream);
        hipMemsetAsync(deg, 0, (size_t)NU * sizeof(float), stream);
        scatter_add<<<scat_blocks, 256, 0, stream>>>(xe, e_idx, u_idx, agg, deg, NEDGE);
        sage_linear_wmma<<<NU / 16, 256, 0, stream>>>(
            agg, deg, xu, user_Wl + off_w, user_bl + off_v, user_Wr + off_w, xnew);
        hipMemsetAsync(stats, 0, 2 * DH * sizeof(float), stream);
        bn_stats<<<(NU + 255) / 256, 128, 0, stream>>>(xnew, stats, NU, 256);
        bn_apply_relu<<<(int)(((long)NU * DH + 255) / 256), 256, 0, stream>>>(
            xnew, stats, user_gamma + off_v, user_beta + off_v, xu, NU);

        // ---------------- event update: messages user -> event ---------------
        hipMemsetAsync(agg, 0, (size_t)NE * DH * sizeof(float), stream);
        hipMemsetAsync(deg, 0, (size_t)NE * sizeof(float), stream);
        scatter_add<<<scat_blocks, 256, 0, stream>>>(xu, u_idx, e_idx, agg, deg, NEDGE);
        sage_linear_wmma<<<NE / 16, 256, 0, stream>>>(
            agg, deg, xe, event_Wl + off_w, event_bl + off_v, event_Wr + off_w, xnew);
        hipMemsetAsync(stats, 0, 2 * DH * sizeof(float), stream);
        bn_stats<<<(NE + 255) / 256, 128, 0, stream>>>(xnew, stats, NE, 256);
        bn_apply_relu<<<(int)(((long)NE * DH + 255) / 256), 256, 0, stream>>>(
            xnew, stats, event_gamma + off_v, event_beta + off_v, xe, NE);
    }
}